// AutoRegressiveDecoderLayer_7842610283509
// MI455X (gfx1250) — compile-verified
//
#include <hip/hip_runtime.h>

// ---------------------------------------------------------------------------
// AutoRegressiveDecoderLayer decode step, MI455X (gfx1250, wave32).
// HBM-bound: ~1.6 GB of KV streamed once -> ~69 us floor at 23.3 TB/s.
// Dense 128x128 linears run on V_WMMA_F32_16X16X4_F32 (one wave per 16x16
// output tile); attention streams K/V with non-temporal b128 loads.
// ---------------------------------------------------------------------------

typedef __attribute__((ext_vector_type(2))) float v2f;
typedef __attribute__((ext_vector_type(4))) float v4f;
typedef __attribute__((ext_vector_type(8))) float v8f;

#define DEV __device__ __forceinline__

constexpr int Bsz   = 1024;
constexpr int E     = 128;
constexpr int H     = 8;
constexpr int HD    = 16;
constexpr int N1    = 1001;   // enc-dec attention length
constexpr int TPREV = 511;    // prior self-attn cache
constexpr int SSA   = 512;    // total self-attn length (TPREV + 1)

DEV v4f ldnt4(const float* p) {           // streaming load, NT hint
    return __builtin_nontemporal_load(reinterpret_cast<const v4f*>(p));
}

DEV float wave_sum(float v) {
#pragma unroll
    for (int m = 16; m >= 1; m >>= 1) v += __shfl_xor(v, m, 32);
    return v;
}
DEV float wave_max(float v) {
#pragma unroll
    for (int m = 16; m >= 1; m >>= 1) v = fmaxf(v, __shfl_xor(v, m, 32));
    return v;
}
DEV v4f wave_sum4(v4f v) {
#pragma unroll
    for (int m = 16; m >= 1; m >>= 1) {
        v.x += __shfl_xor(v.x, m, 32);
        v.y += __shfl_xor(v.y, m, 32);
        v.z += __shfl_xor(v.z, m, 32);
        v.w += __shfl_xor(v.w, m, 32);
    }
    return v;
}
DEV float dot16(v4f a0, v4f a1, v4f a2, v4f a3,
                v4f b0, v4f b1, v4f b2, v4f b3) {
    v4f p = a0 * b0 + a1 * b1 + a2 * b2 + a3 * b3;
    return p.x + p.y + p.z + p.w;
}

// ---------------------------------------------------------------------------
// GEMM: Y[1024,128] = X[1024,128] @ W^T[128,128] + bias (optional ReLU).
// One wave per 16x16 output tile; V_WMMA_F32_16X16X4_F32, K=128 in steps of 4.
// A (16x4 f32, 2 VGPR): lane<16 holds {K=k0,k0+1} of row lane; lane>=16 holds
// {K=k0+2,k0+3} of row lane-16.  B (4x16) mirrors A with W^T (B[k][n]=W[n][k]).
// D (16x16, 8 VGPR): VGPR i = row i (lanes 0-15) / row i+8 (lanes 16-31).
// ---------------------------------------------------------------------------
template <bool RELU>
__global__ void gemm128_bias(const float* __restrict__ X,
                             const float* __restrict__ W,
                             const float* __restrict__ bias,
                             float* __restrict__ Y) {
    const int lane = threadIdx.x;                              // 0..31
    const int tN   = blockIdx.x;                               // 0..7
    const int tM   = blockIdx.y * blockDim.y + threadIdx.y;    // 0..63
    const int r15  = lane & 15;
    const int hi   = lane >> 4;                                // 0 or 1

    const float* xrow = X + (size_t)(tM * 16 + r15) * E;       // A row
    const float* wrow = W + (size_t)(tN * 16 + r15) * E;       // B col (W^T row)

    v8f acc = {};
#pragma unroll
    for (int k0 = 0; k0 < E; k0 += 4) {
        const int k = k0 + hi * 2;
        v2f a; a.x = xrow[k]; a.y = xrow[k + 1];
        v2f b; b.x = wrow[k]; b.y = wrow[k + 1];
        acc = __builtin_amdgcn_wmma_f32_16x16x4_f32(
            false, a, false, b, (short)0, acc, false, false);
    }

    const int col   = tN * 16 + r15;
    const float bn  = bias[col];
    const int rbase = tM * 16 + hi * 8;
#pragma unroll
    for (int i = 0; i < 8; ++i) {
        float v = acc[i] + bn;
        if (RELU) v = fmaxf(v, 0.0f);
        Y[(size_t)(rbase + i) * E + col] = v;
    }
}

// ---------------------------------------------------------------------------
// Self-attention (no mask), S = 512 = 511 cached + 1 fresh (k_sa/v_sa in ws;
// the concat is never materialized). One block per batch row, one wave per
// head. Each lane owns 16 timesteps; softmax via wave32 butterflies.
// ---------------------------------------------------------------------------
__global__ void attn_self(const float* __restrict__ Q,      // h_t [B,128]
                          const float* __restrict__ Kp,     // [B,511,128]
                          const float* __restrict__ Vp,     // [B,511,128]
                          const float* __restrict__ kn,     // [B,128]
                          const float* __restrict__ vn,     // [B,128]
                          float* __restrict__ Out) {        // [B,128]
    const int b    = blockIdx.x;
    const int h    = threadIdx.y;
    const int lane = threadIdx.x;

    const float* qp = Q + (size_t)b * E + h * HD;
    const v4f q0 = *(const v4f*)(qp + 0);
    const v4f q1 = *(const v4f*)(qp + 4);
    const v4f q2 = *(const v4f*)(qp + 8);
    const v4f q3 = *(const v4f*)(qp + 12);

    float s[16];
    float mx = -3.4e38f;
#pragma unroll
    for (int i = 0; i < 16; ++i) {
        const int t = lane + i * 32;
        const float* kp = (t < TPREV)
            ? (Kp + ((size_t)b * TPREV + t) * E + h * HD)
            : (kn + (size_t)b * E + h * HD);
        const v4f k0 = ldnt4(kp + 0),  k1 = ldnt4(kp + 4);
        const v4f k2 = ldnt4(kp + 8),  k3 = ldnt4(kp + 12);
        const float sc = dot16(q0, q1, q2, q3, k0, k1, k2, k3) * 0.25f; // 1/sqrt(16)
        s[i] = sc;
        mx = fmaxf(mx, sc);
    }
    mx = wave_max(mx);

    float sum = 0.f;
#pragma unroll
    for (int i = 0; i < 16; ++i) { s[i] = __expf(s[i] - mx); sum += s[i]; }
    sum = wave_sum(sum);
    const float inv = 1.0f / sum;

    v4f o0 = {}, o1 = {}, o2 = {}, o3 = {};
#pragma unroll
    for (int i = 0; i < 16; ++i) {
        const int t = lane + i * 32;
        const float* vp = (t < TPREV)
            ? (Vp + ((size_t)b * TPREV + t) * E + h * HD)
            : (vn + (size_t)b * E + h * HD);
        const float w = s[i] * inv;
        o0 += w * ldnt4(vp + 0);
        o1 += w * ldnt4(vp + 4);
        o2 += w * ldnt4(vp + 8);
        o3 += w * ldnt4(vp + 12);
    }
    o0 = wave_sum4(o0); o1 = wave_sum4(o1);
    o2 = wave_sum4(o2); o3 = wave_sum4(o3);
    if (lane == 0) {
        float* op = Out + (size_t)b * E + h * HD;
        *(v4f*)(op + 0)  = o0;  *(v4f*)(op + 4)  = o1;
        *(v4f*)(op + 8)  = o2;  *(v4f*)(op + 12) = o3;
    }
}

// ---------------------------------------------------------------------------
// Enc-dec attention with mask (mask true -> score = -1e9). S = 1001, each
// lane owns up to 32 timesteps. Streams ~1.05 GB of K_att/V_att — the main
// bandwidth consumer; all loads are NT b128.
// ---------------------------------------------------------------------------
__global__ void attn_enc(const float* __restrict__ Q,       // [B,128]
                         const float* __restrict__ K,       // [B,1001,128]
                         const float* __restrict__ V,       // [B,1001,128]
                         const unsigned char* __restrict__ mask, // [B,1001] bool
                         float* __restrict__ Out) {         // [B,128]
    const int b    = blockIdx.x;
    const int h    = threadIdx.y;
    const int lane = threadIdx.x;

    const float* qp = Q + (size_t)b * E + h * HD;
    const v4f q0 = *(const v4f*)(qp + 0);
    const v4f q1 = *(const v4f*)(qp + 4);
    const v4f q2 = *(const v4f*)(qp + 8);
    const v4f q3 = *(const v4f*)(qp + 12);

    float s[32];
    float mx = -3.4e38f;
#pragma unroll
    for (int i = 0; i < 32; ++i) {
        const int t = lane + i * 32;
        float sc = -3.4e38f;                       // inactive lanes -> exp = 0
        if (t < N1) {
            const float* kp = K + ((size_t)b * N1 + t) * E + h * HD;
            const v4f k0 = ldnt4(kp + 0),  k1 = ldnt4(kp + 4);
            const v4f k2 = ldnt4(kp + 8),  k3 = ldnt4(kp + 12);
            sc = mask[(size_t)b * N1 + t]
                 ? -1e9f
                 : dot16(q0, q1, q2, q3, k0, k1, k2, k3) * 0.25f;
        }
        s[i] = sc;
        mx = fmaxf(mx, sc);
    }
    mx = wave_max(mx);

    float sum = 0.f;
#pragma unroll
    for (int i = 0; i < 32; ++i) { s[i] = __expf(s[i] - mx); sum += s[i]; }
    sum = wave_sum(sum);
    const float inv = 1.0f / sum;

    v4f o0 = {}, o1 = {}, o2 = {}, o3 = {};
#pragma unroll
    for (int i = 0; i < 32; ++i) {
        const int t = lane + i * 32;
        if (t < N1) {
            const float* vp = V + ((size_t)b * N1 + t) * E + h * HD;
            const float w = s[i] * inv;
            o0 += w * ldnt4(vp + 0);
            o1 += w * ldnt4(vp + 4);
            o2 += w * ldnt4(vp + 8);
            o3 += w * ldnt4(vp + 12);
        }
    }
    o0 = wave_sum4(o0); o1 = wave_sum4(o1);
    o2 = wave_sum4(o2); o3 = wave_sum4(o3);
    if (lane == 0) {
        float* op = Out + (size_t)b * E + h * HD;
        *(v4f*)(op + 0)  = o0;  *(v4f*)(op + 4)  = o1;
        *(v4f*)(op + 8)  = o2;  *(v4f*)(op + 12) = o3;
    }
}

// ---------------------------------------------------------------------------
// out = LayerNorm(R + X) * g + beta, one block (128 threads) per row.
// ---------------------------------------------------------------------------
__global__ void add_ln(const float* __restrict__ R, const float* __restrict__ X,
                       const float* __restrict__ g, const float* __restrict__ be,
                       float* __restrict__ Out) {
    __shared__ float red1[4];
    __shared__ float red2[4];
    const int b = blockIdx.x;
    const int i = threadIdx.x;

    const float x = R[(size_t)b * E + i] + X[(size_t)b * E + i];

    const float p = wave_sum(x);
    if ((i & 31) == 0) red1[i >> 5] = p;
    __syncthreads();
    const float mean = (red1[0] + red1[1] + red1[2] + red1[3]) * (1.0f / E);

    const float d  = x - mean;
    const float p2 = wave_sum(d * d);
    if ((i & 31) == 0) red2[i >> 5] = p2;
    __syncthreads();
    const float var = (red2[0] + red2[1] + red2[2] + red2[3]) * (1.0f / E);

    Out[(size_t)b * E + i] = d * __frsqrt_rn(var + 1e-5f) * g[i] + be[i];
}

// ---------------------------------------------------------------------------
extern "C" void kernel_launch(void* const* d_in, const int* in_sizes, int n_in,
                              void* d_out, int out_size, void* d_ws, size_t ws_size,
                              hipStream_t stream) {
    const float* h_t       = (const float*)d_in[0];
    const float* K_att     = (const float*)d_in[1];
    const float* V_att     = (const float*)d_in[2];
    const float* K_sa_prev = (const float*)d_in[3];
    const float* V_sa_prev = (const float*)d_in[4];
    const unsigned char* mask = (const unsigned char*)d_in[5];
    const float* Wk_w   = (const float*)d_in[6];   const float* Wk_b   = (const float*)d_in[7];
    const float* Wv_w   = (const float*)d_in[8];   const float* Wv_b   = (const float*)d_in[9];
    const float* W0sa_w = (const float*)d_in[10];  const float* W0sa_b = (const float*)d_in[11];
    const float* Wq_w   = (const float*)d_in[12];  const float* Wq_b   = (const float*)d_in[13];
    const float* W0a_w  = (const float*)d_in[14];  const float* W0a_b  = (const float*)d_in[15];
    const float* W1_w   = (const float*)d_in[16];  const float* W1_b   = (const float*)d_in[17];
    const float* W2_w   = (const float*)d_in[18];  const float* W2_b   = (const float*)d_in[19];
    const float* ln_sa_g = (const float*)d_in[20]; const float* ln_sa_b = (const float*)d_in[21];
    const float* ln_ff_g = (const float*)d_in[22]; const float* ln_ff_b = (const float*)d_in[23];

    // workspace: 9 row-buffers of B*E f32 (~4.7 MB total)
    const size_t NBE = (size_t)Bsz * E;
    float* ws    = (float*)d_ws;
    float* k_sa  = ws + 0 * NBE;
    float* v_sa  = ws + 1 * NBE;
    float* a_sa  = ws + 2 * NBE;
    float* t0    = ws + 3 * NBE;
    float* h1    = ws + 4 * NBE;
    float* qb    = ws + 5 * NBE;
    float* a_att = ws + 6 * NBE;
    float* ff1   = ws + 7 * NBE;
    float* h2    = ws + 8 * NBE;

    const dim3 ggrid(E / 16, Bsz / 16 / 4);   // 8 x 16 blocks
    const dim3 gblk(32, 4);                   // 4 waves, one 16x16 tile each
    const dim3 agrid(Bsz);
    const dim3 ablk(32, H);                   // one wave per head

    // KV projection for the new token
    gemm128_bias<false><<<ggrid, gblk, 0, stream>>>(h_t, Wk_w, Wk_b, k_sa);
    gemm128_bias<false><<<ggrid, gblk, 0, stream>>>(h_t, Wv_w, Wv_b, v_sa);

    // self-attention (query = raw h_t, per reference), S = 512
    attn_self<<<agrid, ablk, 0, stream>>>(h_t, K_sa_prev, V_sa_prev, k_sa, v_sa, a_sa);

    // h1 = LN_sa(h_t + W0sa(a_sa))
    gemm128_bias<false><<<ggrid, gblk, 0, stream>>>(a_sa, W0sa_w, W0sa_b, t0);
    add_ln<<<Bsz, E, 0, stream>>>(h_t, t0, ln_sa_g, ln_sa_b, h1);

    // enc-dec attention
    gemm128_bias<false><<<ggrid, gblk, 0, stream>>>(h1, Wq_w, Wq_b, qb);
    attn_enc<<<agrid, ablk, 0, stream>>>(qb, K_att, V_att, mask, a_att);

    // h2 = LN_sa(h1 + W0att(a_att))   (reference reuses ln_sa here)
    gemm128_bias<false><<<ggrid, gblk, 0, stream>>>(a_att, W0a_w, W0a_b, t0);
    add_ln<<<Bsz, E, 0, stream>>>(h1, t0, ln_sa_g, ln_sa_b, h2);

    // FFN: out = LN_ff(h2 + W2(relu(W1(h2))))
    gemm128_bias<true ><<<ggrid, gblk, 0, stream>>>(h2, W1_w, W1_b, ff1);
    gemm128_bias<false><<<ggrid, gblk, 0, stream>>>(ff1, W2_w, W2_b, t0);
    add_ln<<<Bsz, E, 0, stream>>>(h2, t0, ln_ff_g, ln_ff_b, (float*)d_out);
}